// AffineTransformLayer_25108378812765
// MI455X (gfx1250) — compile-verified
//
#include <hip/hip_runtime.h>
#include <cstdint>

// Problem constants (match the reference).
#define B_   4
#define H_   256
#define W_   256
#define C_   64
#define N_   10
#define HM_  64
#define WM_  64
#define INIT_H_ 256
#define INIT_W_ 256
#define PIX_PER_BLK 16   // 16 pixels/block; 16 threads (float4 quads) per pixel

typedef unsigned int u32x4 __attribute__((ext_vector_type(4)));
typedef int          i32x8 __attribute__((ext_vector_type(8)));
typedef int          i32x4 __attribute__((ext_vector_type(4)));

__global__ __launch_bounds__(256)
void affine_warp_max_kernel(const float* __restrict__ x,
                            const float* __restrict__ transforms,
                            const float* __restrict__ mask,
                            float* __restrict__ out)
{
    // TDM-staged mask tile: s_mask[n*WM + xq] = mask[b, n, y>>2, xq]
    __shared__ float s_mask[N_ * WM_];            // 2560 B
    __shared__ int   s_off [N_ * PIX_PER_BLK];    //  640 B  gather offsets (or -1)

    const int b  = blockIdx.z;
    const int y  = blockIdx.y;
    const int x0 = blockIdx.x * PIX_PER_BLK;
    const int t  = threadIdx.x;

    // ---- Phase 0: wave 0 issues the Tensor Data Mover load of the mask row.
    // 2-D tile: 64 contiguous floats (dim0) x 10 transforms (dim1, stride HM*WM).
    if (t == 0) {
        const uint64_t gaddr =
            (uint64_t)(uintptr_t)(mask + (size_t)b * N_ * HM_ * WM_
                                       + (size_t)(y >> 2) * WM_);
        const uint32_t laddr = (uint32_t)(uintptr_t)(&s_mask[0]); // LDS byte addr

        u32x4 g0;
        g0[0] = 1u;                    // count=1 (valid descriptor), user mode
        g0[1] = laddr;                 // lds_addr[31:0]
        g0[2] = (uint32_t)gaddr;       // global_addr[31:0]
        g0[3] = (uint32_t)((gaddr >> 32) & 0x01FFFFFFu) | (2u << 30); // addr[56:32], type=2

        i32x8 g1;
        g1[0] = (2 << 16);                                   // data_size = 4 bytes
        g1[1] = (WM_ & 0xFFFF) << 16;                        // tensor_dim0[15:0]
        g1[2] = ((WM_ >> 16) & 0xFFFF) | ((N_ & 0xFFFF) << 16); // dim0 hi | tensor_dim1 lo
        g1[3] = ((N_ >> 16) & 0xFFFF) | (WM_ << 16);         // dim1 hi | tile_dim0 = 64
        g1[4] = N_;                                          // tile_dim1 = 10, tile_dim2 = 0
        g1[5] = HM_ * WM_;                                   // tensor_dim0_stride (elements)
        g1[6] = 0;
        g1[7] = 0;

        i32x4 gz4 = {0, 0, 0, 0};                            // groups 2/3 unused (2-D)
        i32x8 gz8 = {0, 0, 0, 0, 0, 0, 0, 0};                // extra group (clang-23 form)
        __builtin_amdgcn_tensor_load_to_lds(g0, g1, gz4, gz4, gz8, 0);
    }

    // ---- Phase 1 (overlaps the TDM DMA): 160 threads compute one (n, pixel)
    // projective coordinate each; store gather element-offset (or -1) in LDS.
    if (t < N_ * PIX_PER_BLK) {
        const int n = t >> 4;
        const int p = t & 15;
        const float* tr = transforms + (size_t)(b * N_ + n) * 8;
        const float q0 = tr[0];
        const float q1 = tr[1];
        const float q2 = tr[2] * ((float)H_ / (float)INIT_H_); // p = t / affine_mul
        const float q3 = tr[3];
        const float q4 = tr[4];
        const float q5 = tr[5] * ((float)W_ / (float)INIT_W_);
        const float q6 = tr[6];
        const float q7 = tr[7];

        const float fx = (float)(x0 + p);
        const float fy = (float)y;
        const float k    = q6 * fx + q7 * fy + 1.0f;
        const float in_x = (q0 * fx + q1 * fy + q2) / k;
        const float in_y = (q3 * fx + q4 * fy + q5) / k;
        const int ix = (int)rintf(in_x);   // round-half-to-even, matches jnp.round
        const int iy = (int)rintf(in_y);
        const bool valid = (ix >= 0) && (ix < W_) && (iy >= 0) && (iy < H_);
        const int off = valid ? ((iy * W_ + ix) * C_) : -1;
        s_off[t] = off;
        if (valid) {
            // Warm WGP cache for the phase-2 b128 gathers (global_prefetch_b8).
            __builtin_prefetch(x + (size_t)b * H_ * W_ * C_ + off, 0, 0);
        }
    }

    if (t == 0) {
        __builtin_amdgcn_s_wait_tensorcnt(0);  // mask tile landed in LDS
    }
    __syncthreads();

    // ---- Phase 2: gather + scale + max over n. 16 lanes cover one pixel's
    // 64 channels with float4 -> each gather row is one fully-coalesced 256 B.
    const int pix = t >> 4;
    const int q   = t & 15;
    const int xg  = x0 + pix;
    const int xq  = xg >> 2;
    const float* xb = x + (size_t)b * H_ * W_ * C_ + (q << 2);

    float4 acc;
    acc.x = acc.y = acc.z = acc.w = -__FLT_MAX__;

#pragma unroll
    for (int n = 0; n < N_; ++n) {
        const int   off = s_off[n * PIX_PER_BLK + pix];
        const float mm  = (off >= 0) ? s_mask[n * WM_ + xq] : 0.0f; // invalid -> 0 candidate
        const float4 v  = *reinterpret_cast<const float4*>(xb + (off >= 0 ? off : 0));
        acc.x = fmaxf(acc.x, v.x * mm);
        acc.y = fmaxf(acc.y, v.y * mm);
        acc.z = fmaxf(acc.z, v.z * mm);
        acc.w = fmaxf(acc.w, v.w * mm);
    }

    float* op = out + (((size_t)(b * H_ + y) * W_ + xg) * C_) + (q << 2);
    *reinterpret_cast<float4*>(op) = acc;
}

extern "C" void kernel_launch(void* const* d_in, const int* in_sizes, int n_in,
                              void* d_out, int out_size, void* d_ws, size_t ws_size,
                              hipStream_t stream)
{
    (void)in_sizes; (void)n_in; (void)out_size; (void)d_ws; (void)ws_size;
    const float* x          = (const float*)d_in[0];
    const float* transforms = (const float*)d_in[1];
    const float* mask       = (const float*)d_in[2];
    float* out              = (float*)d_out;

    dim3 grid(W_ / PIX_PER_BLK, H_, B_);   // (16, 256, 4)
    dim3 block(256);                        // 8 waves of 32
    affine_warp_max_kernel<<<grid, block, 0, stream>>>(x, transforms, mask, out);
}